// TokenGenReferenceModule_70360154243473
// MI455X (gfx1250) — compile-verified
//
#include <hip/hip_runtime.h>
#include <hip/hip_bf16.h>

#define NUM_HEADS     64
#define NUM_KV_HEADS  8
#define HEAD_DIM      128
#define Q_MULT        8
#define CACHE_LEN     8192
#define SLIDING_WIN   4096
#define BATCH         16
#define ROW_STRIDE    (NUM_KV_HEADS * HEAD_DIM) // 1024 floats between cache rows
#define WIN_START     (CACHE_LEN - SLIDING_WIN) // 4096 (first row touched)
#define VALID_MIN     (CACHE_LEN + 1 - SLIDING_WIN) // 4097 (first unmasked pos)
#define SM_SCALE      0.08838834764831845f      // 1/sqrt(128)

typedef float v2f __attribute__((ext_vector_type(2)));
typedef float v8f __attribute__((ext_vector_type(8)));

__device__ __forceinline__ v8f wmma_f32_4(v2f a, v2f b, v8f c) {
  // D(16x16,f32) = A(16x4,f32) * B(4x16,f32) + C
  return __builtin_amdgcn_wmma_f32_16x16x4_f32(
      /*neg_a=*/false, a, /*neg_b=*/false, b,
      /*c_mod=*/(short)0, c, /*reuse_a=*/false, /*reuse_b=*/false);
}

// ---------------------------------------------------------------------------
// Kernel 1: split-K flash-decode partial. One wave per (split, kv_head, batch).
// Per 16-row tile:
//   S^T = K_tile x Q^T            (32 chained wmma, K=4 each over head_dim)
//   online softmax per query column (per lane + shfl_xor(16) combine)
//   O^T += V^T x P^T              (8 dim-tiles x 4 chained wmma)
// Writes unnormalized (o, m, l) partials to workspace.
// ---------------------------------------------------------------------------
template<int NS>
__global__ __launch_bounds__(32)
void attn_partial_kernel(const float* __restrict__ Q,
                         const float* __restrict__ Kc,
                         const float* __restrict__ Vc,
                         float* __restrict__ OP,
                         float* __restrict__ Mbuf,
                         float* __restrict__ Lbuf)
{
  constexpr int ROWS_PER_SPLIT  = SLIDING_WIN / NS;
  constexpr int TILES_PER_SPLIT = ROWS_PER_SPLIT / 16;

  const int split = blockIdx.x;
  const int h     = blockIdx.y;
  const int b     = blockIdx.z;
  const int lane  = threadIdx.x;
  const int qcol  = lane & 15;      // query column (real if < 8)
  const int hi    = lane >> 4;      // half-wave selector
  const int off   = hi * 2;         // K-dim sub-offset for A/B layout

  // ---- preload Q^T as B-operands (pre-scaled by 1/sqrt(d)); loop-invariant ----
  v2f qb[32];
  {
    const int qh = (qcol < Q_MULT) ? qcol : 0;
    const float* Qrow = Q + (size_t)b * (NUM_HEADS * HEAD_DIM)
                          + (size_t)(h * Q_MULT + qh) * HEAD_DIM + off;
    const float w = (qcol < Q_MULT) ? SM_SCALE : 0.0f;
#pragma unroll
    for (int kc = 0; kc < 32; ++kc) {
      v2f t = *(const v2f*)(Qrow + 4 * kc);
      qb[kc].x = t.x * w;
      qb[kc].y = t.y * w;
    }
  }

  const int r0 = WIN_START + split * ROWS_PER_SPLIT;
  // K: per-lane row pointer (A layout: lane -> kv row, off -> K-dim pair)
  const float* pK = Kc + ((size_t)(b * CACHE_LEN + r0 + qcol) * NUM_KV_HEADS + h) * HEAD_DIM + off;
  // V: per-lane column pointer (A layout for V^T: lane -> output dim)
  const float* pV = Vc + ((size_t)(b * CACHE_LEN + r0) * NUM_KV_HEADS + h) * HEAD_DIM + qcol;

  v8f acc[8];
#pragma unroll
  for (int t = 0; t < 8; ++t) acc[t] = (v8f){0,0,0,0,0,0,0,0};
  float m = -1e30f, l = 0.0f;

#pragma unroll 1
  for (int it = 0; it < TILES_PER_SPLIT; ++it) {
    const int rbase = r0 + it * 16;

    // ---- S^T = K_tile x Q^T ----
    v8f s = (v8f){0,0,0,0,0,0,0,0};
#pragma unroll
    for (int kc = 0; kc < 32; ++kc) {
      v2f ka = *(const v2f*)(pK + 4 * kc);
      s = wmma_f32_4(ka, qb[kc], s);
    }

    // ---- sliding-window mask (only first tile of the window has masked rows) ----
    if (rbase < VALID_MIN) {
#pragma unroll
      for (int j = 0; j < 8; ++j) {
        const int pos = rbase + j + hi * 8;
        if (pos < VALID_MIN) s[j] = -1e30f;
      }
    }

    // ---- online softmax: lane holds 8 kv-scores of query column qcol ----
    float rm = s[0];
#pragma unroll
    for (int j = 1; j < 8; ++j) rm = fmaxf(rm, s[j]);
    rm = fmaxf(rm, __shfl_xor(rm, 16, 32));
    const float m_new = fmaxf(m, rm);
    const float sc = __expf(m - m_new);

    float p[8];
    float lsum = 0.0f;
#pragma unroll
    for (int j = 0; j < 8; ++j) { p[j] = __expf(s[j] - m_new); lsum += p[j]; }
    lsum += __shfl_xor(lsum, 16, 32);
    l = l * sc + lsum;
    m = m_new;
#pragma unroll
    for (int t = 0; t < 8; ++t) acc[t] *= sc;

    // ---- build P^T B-operands from C layout (half-wave swaps) ----
    float pb0[4], pb1[4];
    {
      const float x0 = __shfl_xor(p[0], 16, 32);
      const float x1 = __shfl_xor(p[1], 16, 32);
      const float x2 = __shfl_xor(p[2], 16, 32);
      const float x3 = __shfl_xor(p[3], 16, 32);
      const float x4 = __shfl_xor(p[4], 16, 32);
      const float x5 = __shfl_xor(p[5], 16, 32);
      const float x6 = __shfl_xor(p[6], 16, 32);
      const float x7 = __shfl_xor(p[7], 16, 32);
      const bool lo = (hi == 0);
      pb0[0] = lo ? p[0] : x2;  pb1[0] = lo ? p[1] : x3;   // kv rows 0..3
      pb0[1] = lo ? p[4] : x6;  pb1[1] = lo ? p[5] : x7;   // kv rows 4..7
      pb0[2] = lo ? x0 : p[2];  pb1[2] = lo ? x1 : p[3];   // kv rows 8..11
      pb0[3] = lo ? x4 : p[6];  pb1[3] = lo ? x5 : p[7];   // kv rows 12..15
    }

    // ---- O^T += V^T x P^T ----
#pragma unroll
    for (int t = 0; t < 8; ++t) {
#pragma unroll
      for (int c = 0; c < 4; ++c) {
        v2f va;
        va.x = pV[(size_t)(4 * c + off)     * ROW_STRIDE + 16 * t];
        va.y = pV[(size_t)(4 * c + off + 1) * ROW_STRIDE + 16 * t];
        v2f pb; pb.x = pb0[c]; pb.y = pb1[c];
        acc[t] = wmma_f32_4(va, pb, acc[t]);
      }
    }

    pK += 16 * ROW_STRIDE;
    pV += 16 * ROW_STRIDE;
  }

  // ---- write partials: lane pair (q, q+16) holds dims {16t+j} / {16t+j+8} ----
  if (qcol < Q_MULT) {
    const size_t pidx = ((size_t)(split * BATCH + b) * NUM_KV_HEADS + h) * Q_MULT + qcol;
    float* op = OP + pidx * HEAD_DIM;
#pragma unroll
    for (int t = 0; t < 8; ++t)
#pragma unroll
      for (int j = 0; j < 8; ++j)
        op[16 * t + j + hi * 8] = acc[t][j];
    if (hi == 0) { Mbuf[pidx] = m; Lbuf[pidx] = l; }
  }
}

// ---------------------------------------------------------------------------
// Kernel 2: merge split partials + new-token term + sink logit, normalize.
// One block per (kv_head, batch); thread = output dim.
// ---------------------------------------------------------------------------
template<int NS>
__global__ __launch_bounds__(HEAD_DIM)
void attn_reduce_kernel(const float* __restrict__ Q,
                        const float* __restrict__ Knew,
                        const float* __restrict__ Vnew,
                        const float* __restrict__ sinks,
                        const float* __restrict__ OP,
                        const float* __restrict__ Mbuf,
                        const float* __restrict__ Lbuf,
                        float* __restrict__ out)
{
  const int h = blockIdx.x;
  const int b = blockIdx.y;
  const int d = threadIdx.x;

  __shared__ float red[HEAD_DIM];
  __shared__ float snew[Q_MULT];
  __shared__ float sm[NS * Q_MULT];
  __shared__ float sl[NS * Q_MULT];

  // s_new[q] = (Q[q] . K_new) * scale   (cooperative LDS reductions)
  const float kn = Knew[(size_t)b * ROW_STRIDE + h * HEAD_DIM + d];
  for (int q = 0; q < Q_MULT; ++q) {
    red[d] = Q[(size_t)b * (NUM_HEADS * HEAD_DIM) + (h * Q_MULT + q) * HEAD_DIM + d] * kn;
    __syncthreads();
    for (int s = HEAD_DIM / 2; s > 0; s >>= 1) {
      if (d < s) red[d] += red[d + s];
      __syncthreads();
    }
    if (d == 0) snew[q] = red[0] * SM_SCALE;
    __syncthreads();
  }

  // stage split (m, l) into LDS
  for (int i = d; i < NS * Q_MULT; i += HEAD_DIM) {
    const int split = i >> 3, q = i & 7;
    const size_t pidx = ((size_t)(split * BATCH + b) * NUM_KV_HEADS + h) * Q_MULT + q;
    sm[i] = Mbuf[pidx];
    sl[i] = Lbuf[pidx];
  }
  __syncthreads();

  const float vn = Vnew[(size_t)b * ROW_STRIDE + h * HEAD_DIM + d];

  for (int q = 0; q < Q_MULT; ++q) {
    const float snk = sinks[h * Q_MULT + q];
    float mg = fmaxf(snew[q], snk);
    for (int s = 0; s < NS; ++s) mg = fmaxf(mg, sm[s * Q_MULT + q]);

    float L = __expf(snk - mg);       // sink contributes to denominator only
    float num = 0.0f;
    for (int s = 0; s < NS; ++s) {
      const float w = __expf(sm[s * Q_MULT + q] - mg);
      L += sl[s * Q_MULT + q] * w;
      const size_t pidx = ((size_t)(s * BATCH + b) * NUM_KV_HEADS + h) * Q_MULT + q;
      num += OP[pidx * HEAD_DIM + d] * w;
    }
    const float wn = __expf(snew[q] - mg);
    L += wn;
    num += wn * vn;

    out[(size_t)b * (NUM_HEADS * HEAD_DIM) + (h * Q_MULT + q) * HEAD_DIM + d] = num / L;
  }
}

// ---------------------------------------------------------------------------
// Launch: pick the largest split count whose workspace fits ws_size.
// Workspace floats per split-count NS: NS*B*KVH*QM*(D + 2)
// ---------------------------------------------------------------------------
template<int NS>
static void launch_impl(const float* Q, const float* K, const float* V,
                        const float* Kc, const float* Vc, const float* sinks,
                        float* out, void* d_ws, hipStream_t stream) {
  float* OP = (float*)d_ws;
  const size_t opN = (size_t)NS * BATCH * NUM_KV_HEADS * Q_MULT * HEAD_DIM;
  float* Mbuf = OP + opN;
  float* Lbuf = Mbuf + (size_t)NS * BATCH * NUM_KV_HEADS * Q_MULT;

  attn_partial_kernel<NS><<<dim3(NS, NUM_KV_HEADS, BATCH), 32, 0, stream>>>(
      Q, Kc, Vc, OP, Mbuf, Lbuf);
  attn_reduce_kernel<NS><<<dim3(NUM_KV_HEADS, BATCH), HEAD_DIM, 0, stream>>>(
      Q, K, V, sinks, OP, Mbuf, Lbuf, out);
}

static inline size_t ws_bytes_needed(int ns) {
  return (size_t)ns * BATCH * NUM_KV_HEADS * Q_MULT * (HEAD_DIM + 2) * sizeof(float);
}

// inputs: 0=Q 1=K 2=V 3=K_cache 4=V_cache 5=sinks  (all f32); out: f32
extern "C" void kernel_launch(void* const* d_in, const int* in_sizes, int n_in,
                              void* d_out, int out_size, void* d_ws, size_t ws_size,
                              hipStream_t stream) {
  (void)in_sizes; (void)n_in; (void)out_size;
  const float* Q     = (const float*)d_in[0];
  const float* K     = (const float*)d_in[1];
  const float* V     = (const float*)d_in[2];
  const float* Kc    = (const float*)d_in[3];
  const float* Vc    = (const float*)d_in[4];
  const float* sinks = (const float*)d_in[5];
  float* out = (float*)d_out;

  if (ws_size >= ws_bytes_needed(32)) {
    launch_impl<32>(Q, K, V, Kc, Vc, sinks, out, d_ws, stream);
  } else if (ws_size >= ws_bytes_needed(16)) {
    launch_impl<16>(Q, K, V, Kc, Vc, sinks, out, d_ws, stream);
  } else if (ws_size >= ws_bytes_needed(8)) {
    launch_impl<8>(Q, K, V, Kc, Vc, sinks, out, d_ws, stream);
  } else if (ws_size >= ws_bytes_needed(4)) {
    launch_impl<4>(Q, K, V, Kc, Vc, sinks, out, d_ws, stream);
  } else {
    launch_impl<1>(Q, K, V, Kc, Vc, sinks, out, d_ws, stream);
  }
}